// GnReluDepthwiseConv_25400436588652
// MI455X (gfx1250) — compile-verified
//
#include <hip/hip_runtime.h>
#include <hip/hip_bf16.h>

// ---- problem constants (match reference) ----
#define V_N      300000
#define C_N      64
#define FE_N     9
#define GROUPS_N 32
#define EPS_N    1e-5f

// ---- tunables ----
#define P1_BLOCKS 256                          // partial rows
#define P1_CHUNK  ((V_N + P1_BLOCKS - 1) / P1_BLOCKS)
#define P3_BLOCKS 2048

typedef float  f4  __attribute__((ext_vector_type(4)));
typedef float  v2f __attribute__((ext_vector_type(2)));
typedef float  v8f __attribute__((ext_vector_type(8)));

// ws layout (floats):
//   [0 .. 256*128)        per-block partials: row b = {sum[0..63], sumsq[0..63]}
//   [256*128 .. +64)      scale[c]
//   [256*128+64 .. +64)   shift[c]
#define WS_PARTIALS 0
#define WS_SCALE    (P1_BLOCKS * 128)
#define WS_SHIFT    (P1_BLOCKS * 128 + 64)

// ---------------------------------------------------------------------------
// Pass 1: per-channel sum / sumsq partials.  256 threads = 16 vertex-rows x
// 16 channel-quads; b128 loads stream lv once (76.8 MB).
// ---------------------------------------------------------------------------
__global__ void gn_partials_kernel(const float* __restrict__ lv,
                                   float* __restrict__ ws) {
    const int tid = threadIdx.x;
    const int c4  = tid & 15;   // channel quad: channels 4*c4 .. 4*c4+3
    const int r   = tid >> 4;   // vertex row slot 0..15

    const int start = blockIdx.x * P1_CHUNK;
    int end = start + P1_CHUNK;
    if (end > V_N) end = V_N;

    const f4* __restrict__ lv4 = (const f4*)lv;

    f4 s = (f4)0.0f;
    f4 q = (f4)0.0f;
    for (int v = start + r; v < end; v += 16) {
        f4 x = lv4[v * 16 + c4];           // global_load_b128, coalesced
        s += x;
        q += x * x;
    }

    __shared__ f4 red[16][32];             // [row][col]: col<16 sum, col>=16 sumsq
    red[r][c4]      = s;
    red[r][16 + c4] = q;
    __syncthreads();

    if (tid < 32) {
        f4 t = (f4)0.0f;
        #pragma unroll
        for (int row = 0; row < 16; ++row) t += red[row][tid];
        f4* ws4 = (f4*)(ws + WS_PARTIALS);
        ws4[blockIdx.x * 32 + tid] = t;    // 128 floats per block, fully written
    }
}

// ---------------------------------------------------------------------------
// Pass 2 (1 block, 8 waves): WMMA reduction of 256 partial rows.
// A = ones(16x4) f32, B = 4 partial rows x 16 cols, D accumulates via C.
// With all-ones A, D[m,n] = sum over the 4 values bound to column n, which is
// invariant to the exact K<->(vgpr,halfwave) mapping -> correct column sums.
// Then fold channel pairs into group stats and emit per-channel scale/shift.
// ---------------------------------------------------------------------------
__global__ void gn_finalize_kernel(const float* __restrict__ gamma,
                                   const float* __restrict__ beta,
                                   float* __restrict__ ws) {
    const int tid  = threadIdx.x;
    const int wave = tid >> 5;            // 0..7 -> 16-column tile each
    const int lane = tid & 31;
    const int col  = wave * 16 + (lane & 15);   // 0..127
    const int hi   = lane >> 4;           // lanes 16..31 take rows p+2, p+3

    const float* __restrict__ part = ws + WS_PARTIALS;

    v2f a; a.x = 1.0f; a.y = 1.0f;        // all-ones A matrix
    v8f acc = {};
    for (int p = 0; p < P1_BLOCKS; p += 4) {
        const int row0 = p + (hi ? 2 : 0);
        v2f b;
        b.x = part[row0 * 128 + col];
        b.y = part[(row0 + 1) * 128 + col];
        acc = __builtin_amdgcn_wmma_f32_16x16x4_f32(
                  false, a, false, b, (short)0, acc, false, false);
    }

    __shared__ float totals[128];         // 0..63 sum[c], 64..127 sumsq[c]
    if ((lane & 31) < 16) totals[col] = acc[0];   // row M=0 (all rows equal)
    __syncthreads();

    if (tid < C_N) {
        const int c = tid;
        const int g = c >> 1;             // 2 channels per group
        const float n    = 2.0f * (float)V_N;
        const float sum  = totals[2 * g] + totals[2 * g + 1];
        const float sq   = totals[64 + 2 * g] + totals[64 + 2 * g + 1];
        const float mean = sum / n;
        float var = sq / n - mean * mean;
        var = fmaxf(var, 0.0f);
        const float inv = rsqrtf(var + EPS_N);
        const float sc  = gamma[c] * inv;
        ws[WS_SCALE + c] = sc;
        ws[WS_SHIFT + c] = beta[c] - mean * sc;
    }
}

// ---------------------------------------------------------------------------
// Pass 3: fused norm+ReLU+depthwise gather conv.
// 256 threads = 16 vertices x 16 channel-quads; each gathered neighbour row is
// read as coalesced b128 (lv stays resident in the 192 MB L2).
// ---------------------------------------------------------------------------
__global__ void conv_kernel(const float* __restrict__ lv,
                            const int* __restrict__ neigh,
                            const float* __restrict__ weight,
                            const float* __restrict__ bias,
                            const float* __restrict__ ws,
                            float* __restrict__ out) {
    const int tid  = threadIdx.x;
    const int c4   = tid & 15;            // channel quad
    const int slot = tid >> 4;            // 0..15 vertices per block-iteration

    __shared__ f4 sW[FE_N][16];
    __shared__ f4 sScale[16];
    __shared__ f4 sShift[16];
    __shared__ f4 sBias[16];

    if (tid < 16) {
        sScale[tid] = ((const f4*)(ws + WS_SCALE))[tid];
        sShift[tid] = ((const f4*)(ws + WS_SHIFT))[tid];
        sBias[tid]  = ((const f4*)bias)[tid];
    }
    if (tid < FE_N * 16) {
        ((f4*)sW)[tid] = ((const f4*)weight)[tid];
    }
    __syncthreads();

    const f4* __restrict__ lv4  = (const f4*)lv;
    f4* __restrict__       out4 = (f4*)out;

    const f4 scl = sScale[c4];
    const f4 sft = sShift[c4];

    for (int vb = blockIdx.x * 16; vb < V_N; vb += gridDim.x * 16) {
        const int v = vb + slot;
        if (v >= V_N) continue;

        f4 acc = sBias[c4];
        const int* nb = neigh + v * FE_N;
        #pragma unroll
        for (int f = 0; f < FE_N; ++f) {
            const int idx = nb[f];                 // uniform within a half-wave
            f4 x = lv4[idx * 16 + c4];             // gathered b128, L2-hit
            f4 xn = x * scl + sft;                 // fused GroupNorm
            xn.x = fmaxf(xn.x, 0.0f);              // ReLU
            xn.y = fmaxf(xn.y, 0.0f);
            xn.z = fmaxf(xn.z, 0.0f);
            xn.w = fmaxf(xn.w, 0.0f);
            acc += sW[f][c4] * xn;                 // depthwise MAC
        }
        out4[v * 16 + c4] = acc;                   // global_store_b128
    }
}

// ---------------------------------------------------------------------------
extern "C" void kernel_launch(void* const* d_in, const int* in_sizes, int n_in,
                              void* d_out, int out_size, void* d_ws, size_t ws_size,
                              hipStream_t stream) {
    const float* lv     = (const float*)d_in[0];
    const int*   neigh  = (const int*)  d_in[1];
    const float* gamma  = (const float*)d_in[2];
    const float* beta   = (const float*)d_in[3];
    const float* weight = (const float*)d_in[4];
    const float* bias   = (const float*)d_in[5];
    float*       out    = (float*)d_out;
    float*       ws     = (float*)d_ws;

    gn_partials_kernel<<<P1_BLOCKS, 256, 0, stream>>>(lv, ws);
    gn_finalize_kernel<<<1, 256, 0, stream>>>(gamma, beta, ws);
    conv_kernel<<<P3_BLOCKS, 256, 0, stream>>>(lv, neigh, weight, bias, ws, out);
}